// LatentVariable_10075993276562
// MI455X (gfx1250) — compile-verified
//
#include <hip/hip_runtime.h>
#include <hip/hip_bf16.h>

#define TPB 256
#define NBLOCKS 1024

typedef float v4f __attribute__((ext_vector_type(4)));

// Stage one lane's 32B eps row into LDS via the CDNA5 async engine (GVS mode:
// saddr = eps base, vaddr = 32-bit byte offset; B*32 = 32MB < 2^32).
__device__ __forceinline__ void async_stage_eps(unsigned ldsAddr, unsigned byteOff,
                                                const float* eps) {
    asm volatile(
        "global_load_async_to_lds_b128 %0, %1, %2 offset:0\n\t"
        "global_load_async_to_lds_b128 %3, %1, %2 offset:16"
        :
        : "v"(ldsAddr), "v"(byteOff), "s"(eps), "v"(ldsAddr + 16u)
        : "memory");
}

// z_b = mu[a_b] + tril(L[a_b]) @ eps_b,  a_b = annotator[b], D = 8.
// Memory/gather-bound (84 MFLOP vs ~90MB HBM): VALU FMAs + async/prefetch
// latency hiding; WMMA is inapplicable (per-element distinct L).
__global__ __launch_bounds__(TPB) void latent_rsample_kernel(
    const int* __restrict__ ann,
    const float* __restrict__ mu,     // [A, 8]
    const float* __restrict__ L,      // [A, 8, 8]
    const float* __restrict__ eps,    // [B, 8]
    float* __restrict__ out,          // [B, 8]
    int B)
{
    __shared__ float sEps[2 * TPB * 8];   // 16 KB: ping-pong, 32B slot per lane

    const int tid = threadIdx.x;
    // Low 32 bits of the generic pointer to a __shared__ object == LDS byte addr.
    const unsigned lds0 = (unsigned)(unsigned long long)(&sEps[tid * 8]);
    const unsigned lds1 = lds0 + TPB * 32u;

    const int t      = blockIdx.x * TPB + tid;
    const int stride = gridDim.x * TPB;
    if (t >= B) return;

    // Prologue: stage first element's eps into buffer 0; load first annotator.
    async_stage_eps(lds0, (unsigned)t * 32u, eps);
    int a = __builtin_nontemporal_load(ann + t);
    int p = 0;

    for (int b = t; b < B; b += stride) {
        // ---- pipeline: stage NEXT element's eps into the other buffer, load
        //      next annotator id, prefetch its mu/L cachelines ----
        int bn  = b + stride;
        int bnc = (bn < B) ? bn : b;                 // clamp: async count constant
        async_stage_eps(p ? lds0 : lds1, (unsigned)bnc * 32u, eps);
        int an = __builtin_nontemporal_load(ann + bnc);
        __builtin_prefetch(L  + (size_t)an * 64,      0, 0);
        __builtin_prefetch(L  + (size_t)an * 64 + 32, 0, 0);
        __builtin_prefetch(mu + (size_t)an * 8,       0, 0);

        // ---- gather mu row + lower-triangular part of L (12 x B128, 192B) ----
        const float4* mu4 = reinterpret_cast<const float4*>(mu) + (size_t)a * 2;
        const float4* L4  = reinterpret_cast<const float4*>(L)  + (size_t)a * 16;
        float4 m0  = mu4[0],  m1  = mu4[1];
        float4 r0  = L4[0];                  // row0 cols 0..3 (only .x used)
        float4 r1  = L4[2];
        float4 r2  = L4[4];
        float4 r3  = L4[6];
        float4 r4a = L4[8],  r4b = L4[9];
        float4 r5a = L4[10], r5b = L4[11];
        float4 r6a = L4[12], r6b = L4[13];
        float4 r7a = L4[14], r7b = L4[15];

        // ---- wait until the CURRENT buffer's 2 async loads have landed:
        //      async completions are in-order per wave; the 2 just issued for
        //      the next element may remain outstanding ----
        asm volatile("s_wait_asynccnt 2" ::: "memory");
        const float4* se =
            reinterpret_cast<const float4*>(&sEps[(p ? TPB * 8 : 0) + tid * 8]);
        float4 e0 = se[0], e1 = se[1];

        // ---- 36 FMAs: strict lower-triangular accumulation ----
        v4f z0, z1;
        z0.x = fmaf(r0.x, e0.x, m0.x);
        z0.y = fmaf(r1.y, e0.y, fmaf(r1.x, e0.x, m0.y));
        z0.z = fmaf(r2.z, e0.z, fmaf(r2.y, e0.y, fmaf(r2.x, e0.x, m0.z)));
        z0.w = fmaf(r3.w, e0.w, fmaf(r3.z, e0.z,
                    fmaf(r3.y, e0.y, fmaf(r3.x, e0.x, m0.w))));
        z1.x = fmaf(r4b.x, e1.x, fmaf(r4a.w, e0.w, fmaf(r4a.z, e0.z,
                    fmaf(r4a.y, e0.y, fmaf(r4a.x, e0.x, m1.x)))));
        z1.y = fmaf(r5b.y, e1.y, fmaf(r5b.x, e1.x, fmaf(r5a.w, e0.w,
                    fmaf(r5a.z, e0.z, fmaf(r5a.y, e0.y,
                    fmaf(r5a.x, e0.x, m1.y))))));
        z1.z = fmaf(r6b.z, e1.z, fmaf(r6b.y, e1.y, fmaf(r6b.x, e1.x,
                    fmaf(r6a.w, e0.w, fmaf(r6a.z, e0.z,
                    fmaf(r6a.y, e0.y, fmaf(r6a.x, e0.x, m1.z)))))));
        z1.w = fmaf(r7b.w, e1.w, fmaf(r7b.z, e1.z, fmaf(r7b.y, e1.y,
                    fmaf(r7b.x, e1.x, fmaf(r7a.w, e0.w,
                    fmaf(r7a.z, e0.z, fmaf(r7a.y, e0.y,
                    fmaf(r7a.x, e0.x, m1.w))))))));

        // ---- coalesced non-temporal B128 stores (don't evict the L2-resident
        //      mu/covtril tables with the 32MB output stream) ----
        v4f* o4 = reinterpret_cast<v4f*>(out) + (size_t)b * 2;
        __builtin_nontemporal_store(z0, o4);
        __builtin_nontemporal_store(z1, o4 + 1);

        a = an;
        p ^= 1;
    }
}

extern "C" void kernel_launch(void* const* d_in, const int* in_sizes, int n_in,
                              void* d_out, int out_size, void* d_ws, size_t ws_size,
                              hipStream_t stream) {
    const int*   ann = (const int*)  d_in[0];   // annotator         [B]
    const float* mu  = (const float*)d_in[1];   // posterior_mu      [A,8]
    const float* L   = (const float*)d_in[2];   // posterior_covtril [A,8,8]
    const float* eps = (const float*)d_in[3];   // eps               [B,8]
    float*       out = (float*)d_out;           // z                 [B,8]
    const int B = in_sizes[0];

    hipLaunchKernelGGL(latent_rsample_kernel, dim3(NBLOCKS), dim3(TPB), 0, stream,
                       ann, mu, L, eps, out, B);
}